// DeepFeatureEmbedding_35064113005003
// MI455X (gfx1250) — compile-verified
//
#include <hip/hip_runtime.h>
#include <stdint.h>

// Problem constants (from the reference)
#define B_   2
#define N_   32768
#define S_   1024
#define CSRC 35
#define K_   32
#define R2_  1.0f
#define COUT 39   // 3 (xyz-norm) + 1 (intensity) + 35 (features)

typedef __attribute__((ext_vector_type(2))) float v2f;
typedef __attribute__((ext_vector_type(8))) float v8f;

static __device__ __forceinline__ unsigned ballot32(bool p) {
    return __builtin_amdgcn_ballot_w32(p);
}

// ---------------------------------------------------------------------------
// Phase 1: ball-query via WMMA distance tiles, 2x unrolled over columns.
// One wave (32 lanes) per 16-keypoint tile. Per iteration: 32 points, two
// v_wmma_f32_16x16x4_f32:
//   A (16x4) : row m = [kx, ky, kz, ||k||^2]    (VGPR0 = K0|K2, VGPR1 = K1|K3)
//   B (4x16) : col n = [-2x, -2y, -2z, 1]^T     (mirrored layout)
//   D[m][n]  = -2 k_m . x_n + ||k_m||^2
// In-radius test:  D[m][n] <= r^2 - ||x_n||^2  (RHS per-lane scalar; lane = N
// in the C/D layout). 16 ballots -> two 16-bit masks per row, merged into one
// 32-bit mask covering columns n0..n0+31; lanes 0..15 pop bits in index order.
// ---------------------------------------------------------------------------
__global__ __launch_bounds__(32) void qbp_wmma_kernel(
    const float* __restrict__ source,     // (B, N, 35) ; xyz = [:, :, :3]
    const float* __restrict__ keypoints,  // (B, S, 4)  ; xyz = [:, :, :3]
    int* __restrict__ idx_out)            // (B, S, K)
{
    const int tile   = blockIdx.x;            // 0 .. B*S/16 - 1
    const int b      = tile / (S_ / 16);
    const int s0     = (tile % (S_ / 16)) * 16;
    const int lane   = (int)threadIdx.x;
    const int lane16 = lane & 15;
    const int hi     = lane >> 4;

    // ---- A tile: keypoint row (both wave halves read the same 16 rows) ----
    const float* kp = keypoints + (((size_t)b * S_) + (size_t)(s0 + lane16)) * 4;
    const float kx = kp[0], ky = kp[1], kz = kp[2];
    const float kn = kx * kx + ky * ky + kz * kz;
    v2f a;
    a.x = hi ? kz : kx;   // VGPR0: K=0 (lanes 0-15) / K=2 (lanes 16-31)
    a.y = hi ? kn : ky;   // VGPR1: K=1 (lanes 0-15) / K=3 (lanes 16-31)

    int          cnt      = 0;
    int          firstIdx = s0 + lane16;      // self index is always in-radius
    const size_t row_base = (((size_t)b * S_) + (size_t)(s0 + lane16)) * K_;
    const float* srcb     = source + (size_t)b * N_ * CSRC;

    for (int n0 = 0; n0 < N_; n0 += 32) {
        // ---- Two B tiles: issue both 16-point gathers before using either ----
        const float* xp0 = srcb + (size_t)(n0 + lane16) * CSRC;
        const float* xp1 = xp0 + 16 * CSRC;
        const float x0 = xp0[0], y0 = xp0[1], z0 = xp0[2];
        const float x1 = xp1[0], y1 = xp1[1], z1 = xp1[2];
        if (n0 + 32 < N_)
            __builtin_prefetch(xp0 + 32 * CSRC, 0, 1);   // global_prefetch_b8

        const float xn0 = x0 * x0 + y0 * y0 + z0 * z0;
        const float xn1 = x1 * x1 + y1 * y1 + z1 * z1;
        v2f bt0, bt1;
        bt0.x = hi ? (-2.0f * z0) : (-2.0f * x0);  // VGPR0: K=0 / K=2
        bt0.y = hi ? 1.0f         : (-2.0f * y0);  // VGPR1: K=1 / K=3
        bt1.x = hi ? (-2.0f * z1) : (-2.0f * x1);
        bt1.y = hi ? 1.0f         : (-2.0f * y1);
        const float t0 = R2_ - xn0;                // per-column scalar (lane = N)
        const float t1 = R2_ - xn1;

        // EXEC is all-ones here (wave-uniform control flow around the WMMAs).
        v8f c0 = {}, c1 = {};
        c0 = __builtin_amdgcn_wmma_f32_16x16x4_f32(
                false, a, false, bt0, (short)0, c0, false, false);
        c1 = __builtin_amdgcn_wmma_f32_16x16x4_f32(
                false, a, false, bt1, (short)0, c1, false, false);

        // VGPR v of C/D holds row M=v (lanes 0-15) and M=v+8 (lanes 16-31).
        unsigned rm0 = 0, rm1 = 0;
        #pragma unroll
        for (int v = 0; v < 8; ++v) {
            const unsigned b0 = ballot32(c0[v] <= t0);
            const unsigned b1 = ballot32(c1[v] <= t1);
            if ((lane & 7) == v) { rm0 = b0; rm1 = b1; }  // handler picks bal[r&7]
        }

        // Lanes 0..15: merge the two 16-bit row masks into one 32-bit mask
        // over columns n0..n0+31 and pop bits in ascending index order.
        if (lane < 16) {
            const unsigned lo = (lane < 8) ? (rm0 & 0xFFFFu) : (rm0 >> 16);
            const unsigned hh = (lane < 8) ? (rm1 & 0xFFFFu) : (rm1 >> 16);
            unsigned m32 = lo | (hh << 16);
            while (m32 && cnt < K_) {
                const int bit  = __builtin_ctz(m32);
                const int gidx = n0 + bit;          // bit 16..31 -> second tile
                if (cnt == 0) firstIdx = gidx;
                idx_out[row_base + cnt] = gidx;
                ++cnt;
                m32 &= m32 - 1;
            }
        }

        // Wave-uniform early exit once every row has K neighbors.
        const unsigned done = ballot32((lane < 16) ? (cnt >= K_) : true);
        if (done == 0xFFFFFFFFu) break;
    }

    // Pad short rows with the first found index (reference semantics).
    if (lane < 16) {
        for (int kk = cnt; kk < K_; ++kk) idx_out[row_base + kk] = firstIdx;
    }
}

// ---------------------------------------------------------------------------
// Phase 2: gather + concat. Pure bandwidth: one thread per output element.
// out[b][s][k][c] = c<3  : xyz[idx][c] - kp[s][c]
//                   c==3 : intensity[idx]
//                   else : source[idx][c-4]
// ---------------------------------------------------------------------------
__global__ __launch_bounds__(256) void gather_kernel(
    const float* __restrict__ source,     // (B, N, 35)
    const float* __restrict__ intensity,  // (B, N, 1)
    const float* __restrict__ keypoints,  // (B, S, 4)
    const int*   __restrict__ idxbuf,     // (B, S, K)
    float*       __restrict__ out)        // (B, S, K, 39)
{
    const long long total = (long long)B_ * S_ * K_ * COUT;
    long long t = (long long)blockIdx.x * blockDim.x + threadIdx.x;
    if (t >= total) return;

    const int c = (int)(t % COUT);
    long long g = t / COUT;
    const int k = (int)(g % K_);
    g /= K_;
    const int s = (int)(g % S_);
    const int b = (int)(g / S_);

    const int idx = idxbuf[(((size_t)b * S_) + s) * K_ + k];
    const float* sp = source + ((size_t)b * N_ + (size_t)idx) * CSRC;

    float val;
    if (c < 3)
        val = sp[c] - keypoints[(((size_t)b * S_) + s) * 4 + c];
    else if (c == 3)
        val = intensity[(size_t)b * N_ + (size_t)idx];
    else
        val = sp[c - 4];
    out[t] = val;
}

extern "C" void kernel_launch(void* const* d_in, const int* in_sizes, int n_in,
                              void* d_out, int out_size, void* d_ws, size_t ws_size,
                              hipStream_t stream) {
    const float* source    = (const float*)d_in[0];  // (B, N, 35)
    const float* intensity = (const float*)d_in[1];  // (B, N, 1)
    const float* keypoints = (const float*)d_in[2];  // (B, S, 4)
    int*   idxbuf = (int*)d_ws;                      // B*S*K ints = 256 KB
    float* out    = (float*)d_out;

    // Phase 1: one wave per 16-keypoint tile -> 128 waves.
    qbp_wmma_kernel<<<dim3(B_ * (S_ / 16)), dim3(32), 0, stream>>>(
        source, keypoints, idxbuf);

    // Phase 2: gather (memory-bound).
    const long long total = (long long)B_ * S_ * K_ * COUT;
    gather_kernel<<<dim3((unsigned)((total + 255) / 256)), dim3(256), 0, stream>>>(
        source, intensity, keypoints, idxbuf, out);
}